// SRNormalization_7232724927090
// MI455X (gfx1250) — compile-verified
//
#include <hip/hip_runtime.h>
#include <stdint.h>

// SRNormalization (LayerNorm w/ Quake fast-inverse-sqrt) for (8,8192,1024) fp32.
// Memory-bound: 512 MB traffic -> ~22us floor @ 23.3 TB/s. No matmul -> no WMMA.
// CDNA5 paths used: async global->LDS double-buffered prefetch (ASYNCcnt),
// b128 vector memory ops, wave32 shfl reductions.

#define HDIM    1024
#define THREADS 256          // 8 waves * 32 lanes; 256 * 4 floats = one row
#define NWAVES  (THREADS / 32)

#if __has_builtin(__builtin_amdgcn_global_load_async_to_lds_b128) && \
    __has_builtin(__builtin_amdgcn_s_wait_asynccnt)
#define USE_ASYNC 1
#else
#define USE_ASYNC 0
#endif

// Exact parameter types of the async-load builtin (from clang diagnostics):
//   param0: int4* in addrspace(1)  (global source)
//   param1: int4* in addrspace(3)  (LDS destination)
typedef int v4i __attribute__((ext_vector_type(4)));
typedef __attribute__((address_space(1))) v4i* as1_v4i;
typedef __attribute__((address_space(3))) v4i* as3_v4i;

__device__ __forceinline__ float fast_inv_sqrt(float v) {
    // Matches reference: bit-trick guess + one Newton iteration, fp32 throughout.
    float x2 = v * 0.5f;
    int   i  = __float_as_int(v);
    i        = 1597463007 - (i >> 1);      // 0x5f3759df
    float y  = __int_as_float(i);
    return y * (1.5f - x2 * y * y);
}

__global__ __launch_bounds__(THREADS) void srnorm_kernel(
    const float* __restrict__ x,
    const float* __restrict__ weight,
    const float* __restrict__ bias,
    float* __restrict__ out,
    int nrows)
{
    __shared__ float xbuf[2][HDIM];        // double-buffered async staging (8 KB)
    __shared__ float red[NWAVES][2];       // per-wave {sum, sumsq}

    const int tid  = threadIdx.x;
    const int wave = tid >> 5;
    const int lane = tid & 31;

    // Row-invariant per-thread weight/bias slice: load once per block (b128).
    const float4 wv = ((const float4*)weight)[tid];
    const float4 bv = ((const float4*)bias)[tid];

    int row = blockIdx.x;
    if (row >= nrows) return;
    const int stride = gridDim.x;

    int cur = 0;
#if USE_ASYNC
    // Prologue: stage first row into buffer 0 (each lane moves 16 B; one
    // async instruction per wave covers that wave's 512 B slice).
    __builtin_amdgcn_global_load_async_to_lds_b128(
        (as1_v4i)(x + (size_t)row * HDIM + tid * 4),
        (as3_v4i)(&xbuf[0][tid * 4]), 0, 0);
#endif

    for (; row < nrows; row += stride) {
        float4 xv;
#if USE_ASYNC
        // Each wave waits only for its own ASYNCcnt; it reads only the LDS
        // bytes its own lanes staged, so no barrier is needed for x data.
        __builtin_amdgcn_s_wait_asynccnt(0);
        xv = *(const float4*)(&xbuf[cur][tid * 4]);      // ds_load_b128

        // Prefetch next row into the other buffer while we compute.
        int next = row + stride;
        if (next >= nrows) next = row;                   // uniform dummy refetch
        __builtin_amdgcn_global_load_async_to_lds_b128(
            (as1_v4i)(x + (size_t)next * HDIM + tid * 4),
            (as3_v4i)(&xbuf[cur ^ 1][tid * 4]), 0, 0);
#else
        xv = ((const float4*)(x + (size_t)row * HDIM))[tid];  // global_load_b128
#endif
        // Per-thread partial sums over 4 elements.
        float s = (xv.x + xv.y) + (xv.z + xv.w);
        float q = fmaf(xv.x, xv.x, fmaf(xv.y, xv.y, fmaf(xv.z, xv.z, xv.w * xv.w)));

        // wave32 butterfly reduction (5 steps).
        #pragma unroll
        for (int m = 16; m >= 1; m >>= 1) {
            s += __shfl_xor(s, m, 32);
            q += __shfl_xor(q, m, 32);
        }
        if (lane == 0) { red[wave][0] = s; red[wave][1] = q; }
        __syncthreads();
        float sum = 0.0f, sq = 0.0f;
        #pragma unroll
        for (int w = 0; w < NWAVES; ++w) { sum += red[w][0]; sq += red[w][1]; }
        __syncthreads();   // protect red[] before next iteration overwrites it

        const float inv_h = 1.0f / (float)HDIM;
        float mean = sum * inv_h;
        float var  = fmaf(sq, inv_h, -(mean * mean));   // E[x^2] - E[x]^2
        float r    = fast_inv_sqrt(var + 1e-5f);

        float4 o;
        o.x = fmaf(wv.x, (xv.x - mean) * r, bv.x);
        o.y = fmaf(wv.y, (xv.y - mean) * r, bv.y);
        o.z = fmaf(wv.z, (xv.z - mean) * r, bv.z);
        o.w = fmaf(wv.w, (xv.w - mean) * r, bv.w);
        ((float4*)(out + (size_t)row * HDIM))[tid] = o;  // global_store_b128

        cur ^= 1;
    }
}

extern "C" void kernel_launch(void* const* d_in, const int* in_sizes, int n_in,
                              void* d_out, int out_size, void* d_ws, size_t ws_size,
                              hipStream_t stream)
{
    const float* x      = (const float*)d_in[0];
    const float* weight = (const float*)d_in[1];
    const float* bias   = (const float*)d_in[2];
    float*       out    = (float*)d_out;

    const int nrows = in_sizes[0] / HDIM;   // 8 * 8192 = 65536

    // Persistent blocks: ~8 rows per block so the async double-buffer pipeline
    // has depth; 8192 blocks * 8 waves saturates the device.
    int blocks = nrows < 8192 ? nrows : 8192;

    srnorm_kernel<<<blocks, THREADS, 0, stream>>>(x, weight, bias, out, nrows);
}